// SGC_74869869904022
// MI455X (gfx1250) — compile-verified
//
#include <hip/hip_runtime.h>

typedef __attribute__((ext_vector_type(2))) float v2f;
typedef __attribute__((ext_vector_type(8))) float v8f;

#define N_NODES 50000
#define N_EDGES 800000
#define N_HOPS  3
#define D_IN    128
#define D_OUT   128
#define K_TOT   (N_HOPS * D_IN)   // 384
#define LDS_STRIDE (K_TOT + 4)    // 388: breaks 64-bank conflict (384 % 64 == 0)

// ---------------------------------------------------------------------------
// Phase 0: zero the hop accumulator h [N_NODES, 384] in workspace
// ---------------------------------------------------------------------------
__global__ void sgc_zero_kernel(float4* __restrict__ h4, long n4) {
    long i = (long)blockIdx.x * blockDim.x + threadIdx.x;
    long stride = (long)gridDim.x * blockDim.x;
    float4 z; z.x = 0.f; z.y = 0.f; z.z = 0.f; z.w = 0.f;
    for (; i < n4; i += stride) h4[i] = z;
}

// ---------------------------------------------------------------------------
// Phase 1: fused 3-hop weighted scatter-add.
// One wave (32 lanes) per edge. Lane loads x[src] as float4 (coalesced 512B
// row read -> one global_load_b128 per lane), scales by w, and does 4
// global_atomic_add_f32 into h[dst, hop*128 ..]. x and h are L2-resident.
// ---------------------------------------------------------------------------
__global__ void sgc_scatter_kernel(const float* __restrict__ x,
                                   const int*   __restrict__ edge_src,
                                   const int*   __restrict__ edge_dst,
                                   const float* __restrict__ edge_w,
                                   float*       __restrict__ h) {
    int gid  = blockIdx.x * blockDim.x + threadIdx.x;
    int wave = gid >> 5;               // one wave per (hop, edge) pair
    int lane = threadIdx.x & 31;
    if (wave >= N_HOPS * N_EDGES) return;

    int hop = wave / N_EDGES;          // edge arrays are [3, E] flat

    int   s  = edge_src[wave];
    int   d  = edge_dst[wave];
    float wt = edge_w[wave];

    const float4* xs = (const float4*)(x + (size_t)s * D_IN);
    float4 v = xs[lane];               // coalesced: 32 lanes cover the 512B row

    float* hd = h + (size_t)d * K_TOT + hop * D_IN + lane * 4;
    atomicAdd(hd + 0, v.x * wt);
    atomicAdd(hd + 1, v.y * wt);
    atomicAdd(hd + 2, v.z * wt);
    atomicAdd(hd + 3, v.w * wt);
}

// ---------------------------------------------------------------------------
// Phase 2: out = h @ W + b with V_WMMA_F32_16X16X4_F32 (exact f32 path).
// Block = 256 threads = 8 waves. Block owns a 16-row tile of h (staged in
// LDS, padded stride); wave w owns output cols [16w, 16w+16). K-loop: 96
// WMMAs over K=384.
//
// Fragment layouts (ISA 7.12.2):
//   A 16x4 f32 : lane l -> row = l&15, khalf = l>>4; VGPR0 = K(2*khalf),
//                VGPR1 = K(2*khalf+1)
//   B 4x16 f32 : lane l -> col = l&15, khalf = l>>4; VGPR0/1 = rows as above
//   C/D 16x16  : VGPR r, lane l -> row = r + (l>=16 ? 8 : 0), col = l&15
// ---------------------------------------------------------------------------
__global__ void sgc_gemm_kernel(const float* __restrict__ h,
                                const float* __restrict__ W,
                                const float* __restrict__ bvec,
                                float*       __restrict__ out) {
    __shared__ float As[16 * LDS_STRIDE];   // 16 x 388 floats = 24832 B

    const int rowTile = blockIdx.x * 16;

    // Cooperative staged load of the 16x384 A tile (float4, coalesced).
    for (int i = threadIdx.x; i < 16 * (K_TOT / 4); i += 256) {
        int r  = i / (K_TOT / 4);
        int c4 = i % (K_TOT / 4);
        float4 v = ((const float4*)(h + (size_t)(rowTile + r) * K_TOT))[c4];
        ((float4*)(As + r * LDS_STRIDE))[c4] = v;   // row base 1552B: 16B aligned
    }
    __syncthreads();

    const int wave  = threadIdx.x >> 5;        // col tile 0..7
    const int lane  = threadIdx.x & 31;
    const int halfK = (lane >> 4) << 1;        // 0 or 2
    const int arow  = lane & 15;
    const int col   = (wave << 4) + (lane & 15);

    v8f c = {};

    #pragma unroll 4
    for (int k = 0; k < K_TOT; k += 4) {
        v2f a, bm;
        const float* ap = As + arow * LDS_STRIDE + k + halfK;   // ds_load_b64
        a.x = ap[0];
        a.y = ap[1];
        bm.x = W[(size_t)(k + halfK)     * D_OUT + col];
        bm.y = W[(size_t)(k + halfK + 1) * D_OUT + col];
        // 8 args: (neg_a, A, neg_b, B, c_mod, C, reuse_a, reuse_b)
        c = __builtin_amdgcn_wmma_f32_16x16x4_f32(
                false, a, false, bm, (short)0, c, false, false);
    }

    const float bias    = bvec[col];
    const int   rowBase = rowTile + ((lane >> 4) << 3);  // +8 for upper half
    #pragma unroll
    for (int r = 0; r < 8; ++r) {
        out[(size_t)(rowBase + r) * D_OUT + col] = c[r] + bias;
    }
}

// ---------------------------------------------------------------------------
extern "C" void kernel_launch(void* const* d_in, const int* in_sizes, int n_in,
                              void* d_out, int out_size, void* d_ws, size_t ws_size,
                              hipStream_t stream) {
    const float* x        = (const float*)d_in[0];
    const int*   edge_src = (const int*)  d_in[1];
    const int*   edge_dst = (const int*)  d_in[2];
    const float* edge_w   = (const float*)d_in[3];
    const float* W        = (const float*)d_in[4];
    const float* bvec     = (const float*)d_in[5];
    float*       out      = (float*)d_out;
    float*       h        = (float*)d_ws;          // N_NODES * 384 f32 = 76.8 MB

    // Phase 0: zero h
    long n4 = (long)N_NODES * K_TOT / 4;
    sgc_zero_kernel<<<2048, 256, 0, stream>>>((float4*)h, n4);

    // Phase 1: fused 3-hop scatter-add (one wave per edge)
    long totalWaves = (long)N_HOPS * N_EDGES;          // 2.4M waves
    long scatterBlocks = (totalWaves + 7) / 8;         // 8 waves per 256-thr block
    sgc_scatter_kernel<<<(int)scatterBlocks, 256, 0, stream>>>(
        x, edge_src, edge_dst, edge_w, h);

    // Phase 2: WMMA GEMM + bias
    sgc_gemm_kernel<<<N_NODES / 16, 256, 0, stream>>>(h, W, bvec, out);
}